// _ScaledDotProductAttention_28870770164471
// MI455X (gfx1250) — compile-verified
//
#include <hip/hip_runtime.h>
#include <stdint.h>

typedef __attribute__((ext_vector_type(2))) float v2f;
typedef __attribute__((ext_vector_type(8))) float v8f;

constexpr int kS = 2048;      // sequence length
constexpr int kD = 64;        // head dim
constexpr int kBH = 32;       // B*H
constexpr int kWaves = 8;     // waves per block
constexpr float kMaskFill = -1.0e9f;

__device__ __forceinline__ v8f wmma_f32(v2f a, v2f b, v8f c) {
  // D = A(16x4 f32) * B(4x16 f32) + C(16x16 f32)
  return __builtin_amdgcn_wmma_f32_16x16x4_f32(
      /*neg_a=*/false, a, /*neg_b=*/false, b,
      /*c_mod=*/(short)0, c, /*reuse_a=*/false, /*reuse_b=*/false);
}

// Compute one 16x16 masked, scaled score tile (rows m0..m0+15, cols n0..n0+15).
__device__ __forceinline__ v8f score_tile(const float* __restrict__ kp,
                                          const uint8_t* __restrict__ mp,
                                          const v2f* aq, int n0, int lane, int m0) {
  const int nIdx = lane & 15;          // N / column-in-tile
  const int koff = (lane >> 4) << 1;   // K sub-pair select
  v8f c = {0.f, 0.f, 0.f, 0.f, 0.f, 0.f, 0.f, 0.f};
#pragma unroll
  for (int kk = 0; kk < 16; ++kk) {    // K = 64 in steps of 4
    const int kb = kk * 4 + koff;
    v2f b;
    b.x = kp[(size_t)kb * kS + n0 + nIdx];        // K is [D, S] (pre-transposed)
    b.y = kp[(size_t)(kb + 1) * kS + n0 + nIdx];
    c = wmma_f32(aq[kk], b, c);
  }
  // scale by 1/sqrt(64), then mask (mask==true -> -1e9), in C layout:
  // VGPR r, lane l -> row m0 + r + 8*(l/16), col n0 + (l&15)
  const size_t mbase = (size_t)(m0 + ((lane >> 4) << 3)) * kS + n0 + nIdx;
#pragma unroll
  for (int r = 0; r < 8; ++r) {
    float s = c[r] * 0.125f;
    if (mp[mbase + (size_t)r * kS]) s = kMaskFill;
    c[r] = s;
  }
  return c;
}

__global__ void __launch_bounds__(256, 1)
attn_f32_wmma_kernel(const float* __restrict__ q, const float* __restrict__ k,
                     const float* __restrict__ v, const uint8_t* __restrict__ mask,
                     float* __restrict__ ctx_out, float* __restrict__ attn_out) {
  __shared__ float plds[kWaves * 256];   // per-wave 16x16 P-tile scratch

  const int tid = threadIdx.x;
  const int wave = tid >> 5;
  const int lane = tid & 31;
  const int bh = blockIdx.x >> 4;        // 16 m-blocks per (b,h)
  const int mblk = blockIdx.x & 15;
  const int m0 = mblk * (16 * kWaves) + wave * 16;

  const float* qp = q + (size_t)bh * kS * kD;
  const float* kp = k + (size_t)bh * kD * kS;
  const float* vp = v + (size_t)bh * kS * kD;
  const uint8_t* mp = mask + (size_t)bh * kS * kS;
  float* ap_out = attn_out + (size_t)bh * kS * kS;
  float* cp_out = ctx_out + (size_t)bh * kS * kD;

  const int nIdx = lane & 15;
  const int koff = (lane >> 4) << 1;

  // ---- Load Q fragments once (A-matrix layout 16x4 per step, 16 steps) ----
  v2f aq[16];
  {
    const float* qrow = qp + (size_t)(m0 + nIdx) * kD;
#pragma unroll
    for (int kk = 0; kk < 16; ++kk) {
      aq[kk].x = qrow[kk * 4 + koff];
      aq[kk].y = qrow[kk * 4 + koff + 1];
    }
  }

  float runmax[8], runsum[8];
#pragma unroll
  for (int r = 0; r < 8; ++r) { runmax[r] = -INFINITY; runsum[r] = 0.f; }

  // ---- Pass 1: online row-max and sum(exp) over all column tiles ----
  for (int nt = 0; nt < kS / 16; ++nt) {
    const int n0 = nt * 16;
    // prefetch next K tile columns (streaming hint)
    __builtin_prefetch(&kp[(size_t)koff * kS + n0 + 16 + nIdx], 0, 1);
    v8f c = score_tile(kp, mp, aq, n0, lane, m0);
#pragma unroll
    for (int r = 0; r < 8; ++r) {
      float tmax = c[r];
#pragma unroll
      for (int off = 1; off < 16; off <<= 1)
        tmax = fmaxf(tmax, __shfl_xor(tmax, off, 32));
      const float nmax = fmaxf(runmax[r], tmax);
      float p = __expf(c[r] - nmax);
#pragma unroll
      for (int off = 1; off < 16; off <<= 1)
        p += __shfl_xor(p, off, 32);
      runsum[r] = runsum[r] * __expf(runmax[r] - nmax) + p;
      runmax[r] = nmax;
    }
  }

  float invs[8];
#pragma unroll
  for (int r = 0; r < 8; ++r) invs[r] = 1.0f / runsum[r];

  v8f ctx[4];
#pragma unroll
  for (int nd = 0; nd < 4; ++nd) {
    v8f z = {0.f, 0.f, 0.f, 0.f, 0.f, 0.f, 0.f, 0.f};
    ctx[nd] = z;
  }

  float* my = plds + wave * 256;

  // ---- Pass 2: recompute scores, normalize, emit attn, accumulate P@V ----
  for (int nt = 0; nt < kS / 16; ++nt) {
    const int n0 = nt * 16;
    v8f c = score_tile(kp, mp, aq, n0, lane, m0);
    float pr[8];
    const size_t abase = (size_t)(m0 + ((lane >> 4) << 3)) * kS + n0 + nIdx;
#pragma unroll
    for (int r = 0; r < 8; ++r) {
      pr[r] = __expf(c[r] - runmax[r]) * invs[r];
      // attn is streamed once, never reused: non-temporal store
      __builtin_nontemporal_store(pr[r], ap_out + abase + (size_t)r * kS);
      my[(r + ((lane >> 4) << 3)) * 16 + nIdx] = pr[r];  // C-layout -> LDS
    }
    __syncthreads();  // make P tile visible across lanes (uniform across waves)
    // Re-read P in A-matrix layout for P(16x16) @ V(16x64)
    v2f ap[4];
#pragma unroll
    for (int kk = 0; kk < 4; ++kk) {
      const int kb = kk * 4 + koff;
      ap[kk].x = my[nIdx * 16 + kb];
      ap[kk].y = my[nIdx * 16 + kb + 1];
    }
    __syncthreads();  // protect LDS against next iteration's overwrite
#pragma unroll
    for (int nd = 0; nd < 4; ++nd) {
#pragma unroll
      for (int kk = 0; kk < 4; ++kk) {
        const int kb = kk * 4 + koff;
        v2f b;
        b.x = vp[(size_t)(n0 + kb) * kD + nd * 16 + nIdx];      // V is [S, D]
        b.y = vp[(size_t)(n0 + kb + 1) * kD + nd * 16 + nIdx];
        ctx[nd] = wmma_f32(ap[kk], b, ctx[nd]);
      }
    }
  }

  // ---- Store context: C layout -> [S, D] ----
  const size_t cbase = (size_t)(m0 + ((lane >> 4) << 3)) * kD + nIdx;
#pragma unroll
  for (int nd = 0; nd < 4; ++nd) {
#pragma unroll
    for (int r = 0; r < 8; ++r) {
      cp_out[cbase + (size_t)r * kD + nd * 16] = ctx[nd][r];
    }
  }
}

extern "C" void kernel_launch(void* const* d_in, const int* in_sizes, int n_in,
                              void* d_out, int out_size, void* d_ws, size_t ws_size,
                              hipStream_t stream) {
  const float* q = (const float*)d_in[0];
  const float* k = (const float*)d_in[1];       // [B,H,D,S] pre-transposed
  const float* v = (const float*)d_in[2];
  const uint8_t* mask = (const uint8_t*)d_in[3];
  float* ctx = (float*)d_out;                                   // [B,H,S,D]
  float* attn = ctx + (size_t)kBH * kS * kD;                    // [B,H,S,S]
  dim3 grid(kBH * (kS / (16 * kWaves)));  // 32 heads * 16 m-blocks = 512
  dim3 block(256);                        // 8 waves, one 16-row tile each
  attn_f32_wmma_kernel<<<grid, block, 0, stream>>>(q, k, v, mask, ctx, attn);
}